// Random_frequency_replacing_16784732192922
// MI455X (gfx1250) — compile-verified
//
#include <hip/hip_runtime.h>
#include <hip/hip_bf16.h>

typedef __attribute__((ext_vector_type(2))) float v2f;
typedef __attribute__((ext_vector_type(8))) float v8f;

#define HH     512
#define WW     512
#define HWPIX  (HH * WW)          // 262144 pixels
#define NB_    100                // histogram bins
#define BATCH  16
#define NCHUNK 25                 // 100 bins / K=4 per WMMA
#define MAXR   362.03867196751236f // sqrt(512^2+512^2)/2 = 256*sqrt(2)
#define LOG2E  1.44269504088896340736f

// sigmoid(r - d), minimizing quarter-rate transcendental ops.
#if __has_builtin(__builtin_amdgcn_tanhf)
// sigmoid(y) = 0.5 + 0.5*tanh(y/2):  v_fma + v_tanh_f32 + v_fma  (1 trans op)
__device__ __forceinline__ float fast_sigmoid(float dist, float r_half) {
  float t = __builtin_amdgcn_tanhf(__builtin_fmaf(dist, -0.5f, r_half));
  return __builtin_fmaf(t, 0.5f, 0.5f);
}
#define RSCALE 0.5f               // pass r/2
#else
// sigmoid(y) = 1/(1 + 2^(-y*log2e)): v_fma + v_exp + v_add + v_rcp (2 trans ops)
__device__ __forceinline__ float fast_sigmoid(float dist, float r_log2e) {
  float e = __builtin_amdgcn_exp2f(__builtin_fmaf(dist, LOG2E, -r_log2e));
  return __builtin_amdgcn_rcpf(1.0f + e);
}
#define RSCALE LOG2E              // pass r*log2e
#endif

// One wave computes D[16x16] = W[16x100] * S[100x16]:
//   M = batch (16), N = 16 consecutive pixels, K = bins (25 chunks of 4)
//   W[b,k] = v[b,k] - v[b,k+1]  (v[b,100] == 0, telescoped sigmoid-diff masks)
//   S[k,p] = sigmoid((k+1)*0.01*maxR - dist(p))  computed on the fly.
__global__ __launch_bounds__(256) void fq_mask_wmma(const float* __restrict__ vs,
                                                    float* __restrict__ out,
                                                    int num_tiles) {
  const int lane        = threadIdx.x & 31;
  const int waveInBlk   = threadIdx.x >> 5;
  const int wavesPerBlk = blockDim.x >> 5;
  const int waveId      = blockIdx.x * wavesPerBlk + waveInBlk;
  const int totalWaves  = gridDim.x * wavesPerBlk;

  const int  m    = lane & 15;        // A: batch row M; B: pixel column N
  const bool hi   = lane >= 16;
  const int  koff = hi ? 2 : 0;       // lanes 0-15 hold K={0,1}, lanes 16-31 K={2,3}

  // ---- Load A weights once per wave (reused across all pixel tiles) ----
  // f32 16x4 A layout: lane m holds VGPR0=K(koff), VGPR1=K(koff+1) per chunk.
  float aw0[NCHUNK], aw1[NCHUNK];
  // Pre-scaled radii for this lane's two K slots per chunk (loop-invariant).
  float rs0[NCHUNK], rs1[NCHUNK];
  const float* __restrict__ vrow = vs + m * NB_;
#pragma unroll
  for (int c = 0; c < NCHUNK; ++c) {
    int   k0 = c * 4 + koff;
    float v0 = vrow[k0];
    float v1 = vrow[k0 + 1];
    float v2 = (k0 + 2 < NB_) ? vrow[k0 + 2] : 0.0f;  // v[b,100] == 0
    aw0[c] = v0 - v1;
    aw1[c] = v1 - v2;
    rs0[c] = (float)(k0 + 1) * (0.01f * MAXR) * RSCALE;
    rs1[c] = (float)(k0 + 2) * (0.01f * MAXR) * RSCALE;
  }

  for (int tile = waveId; tile < num_tiles; tile += totalWaves) {
    // 16 consecutive pixels (one image row segment); this lane's pixel = base+m
    int   p0   = tile * 16;
    int   row  = p0 >> 9;                 // / 512
    int   col  = (p0 & (WW - 1)) + m;
    float fy   = (float)row - 256.0f;     // (a - H/2)
    float fx   = (float)col - 256.0f;     // (b - W/2)
    float dist = __builtin_amdgcn_sqrtf(fy * fy + fx * fx);  // v_sqrt_f32

    v8f acc = {0.f, 0.f, 0.f, 0.f, 0.f, 0.f, 0.f, 0.f};
#pragma unroll
    for (int c = 0; c < NCHUNK; ++c) {
      float s0 = fast_sigmoid(dist, rs0[c]);
      float s1 = fast_sigmoid(dist, rs1[c]);

      v2f a; a.x = aw0[c]; a.y = aw1[c];
      v2f b; b.x = s0;     b.y = s1;
      // 8 args: (neg_a, A, neg_b, B, c_mod, C, reuse_a, reuse_b)
      acc = __builtin_amdgcn_wmma_f32_16x16x4_f32(
          /*neg_a=*/false, a, /*neg_b=*/false, b,
          /*c_mod=*/(short)0, acc, /*reuse_a=*/false, /*reuse_b=*/false);
    }

    // D 16x16 f32: lanes 0-15 -> M = vg (batches 0-7), lanes 16-31 -> M = vg+8
#pragma unroll
    for (int vg = 0; vg < 8; ++vg) {
      int bm = vg + (hi ? 8 : 0);
      out[(size_t)bm * HWPIX + (size_t)(p0 + m)] = acc[vg];
    }
  }
}

// Second tuple element: pass value_set through to the output tail.
__global__ void copy_value_set(const float* __restrict__ vs, float* __restrict__ out, int n) {
  int i = blockIdx.x * blockDim.x + threadIdx.x;
  if (i < n) out[i] = vs[i];
}

extern "C" void kernel_launch(void* const* d_in, const int* in_sizes, int n_in,
                              void* d_out, int out_size, void* d_ws, size_t ws_size,
                              hipStream_t stream) {
  (void)in_sizes; (void)n_in; (void)out_size; (void)d_ws; (void)ws_size;
  // d_in[0] = x [16,3,512,512] f32 : only its shape matters -> unused.
  const float* vs  = (const float*)d_in[1];   // value_set [16,100] f32
  float*       out = (float*)d_out;           // [16*512*512] fq_mask ++ [1600] value_set

  const int num_tiles = HWPIX / 16;           // 16384 pixel tiles
  // 256 blocks x 8 waves = 2048 waves -> exactly 8 tiles per wave.
  fq_mask_wmma<<<dim3(256), dim3(256), 0, stream>>>(vs, out, num_tiles);

  const int nvs = BATCH * NB_;                // 1600
  copy_value_set<<<dim3((nvs + 255) / 256), dim3(256), 0, stream>>>(
      vs, out + (size_t)BATCH * HWPIX, nvs);
}